// UncertainClusterMemory_2473901163211
// MI455X (gfx1250) — compile-verified
//
#include <hip/hip_runtime.h>
#include <math.h>

// ---------------------------------------------------------------------------
// UncertainClusterMemory for MI455X (gfx1250)
// B=64, N=65536, D=1024, U=16. Dominant cost is streaming features (256MB)
// through the WMMA GEMM plus the 512MB feature copy -> bandwidth bound.
// GEMM uses V_WMMA_F32_16X16X4_F32 (exact f32; logits feed softmax at T=0.05
// so f16 quantization would be risky, and the GEMM is memory-bound anyway).
// ---------------------------------------------------------------------------

typedef __attribute__((ext_vector_type(2))) float v2f;
typedef __attribute__((ext_vector_type(8))) float v8f;

#define BB 64
#define NN 65536
#define DD 1024
#define UU 16
#define T_INV 20.0f   // 1/TEMP,    TEMP=0.05
#define U_INV 2.0f    // 1/UM_TEMP, UM_TEMP=0.5

// ---------------- shared-memory block reductions (fixed order => deterministic)
__device__ __forceinline__ float blk_sum(float v, float* red) {
  int t = threadIdx.x;
  red[t] = v; __syncthreads();
  for (int o = blockDim.x >> 1; o > 0; o >>= 1) {
    if (t < o) red[t] += red[t + o];
    __syncthreads();
  }
  float r = red[0]; __syncthreads();
  return r;
}
__device__ __forceinline__ float blk_max(float v, float* red) {
  int t = threadIdx.x;
  red[t] = v; __syncthreads();
  for (int o = blockDim.x >> 1; o > 0; o >>= 1) {
    if (t < o) red[t] = fmaxf(red[t], red[t + o]);
    __syncthreads();
  }
  float r = red[0]; __syncthreads();
  return r;
}

// ---------------- K0: x = inputs / max(||inputs||, eps), per row -------------
__global__ void k_norm_x(const float* __restrict__ in, float* __restrict__ xn) {
  __shared__ float red[256];
  int b = blockIdx.x, t = threadIdx.x;
  const float* row = in + (size_t)b * DD;
  float s = 0.f;
  for (int i = t; i < DD; i += 256) { float v = row[i]; s += v * v; }
  float ss = blk_sum(s, red);
  float inv = 1.0f / fmaxf(sqrtf(ss), 1e-12f);
  for (int i = t; i < DD; i += 256) xn[(size_t)b * DD + i] = row[i] * inv;
}

// ---------------- K1: logits = x @ features^T via V_WMMA_F32_16X16X4_F32 -----
// One wave computes a 64(M) x 16(N) strip: 4 accumulators, features (B frag)
// streamed from HBM exactly once, x (A frags) cache-resident (256KB).
// f32 fragment layout (ISA 7.12.2): lane l holds elements [2*(l>>4), 2*(l>>4)+1]
// of row (l&15) => one 8-byte load per matrix per K=4 step.
// K loop split: main phase with unconditional in-bounds prefetch, tail without,
// so the steady-state loop has no prefetch branches/SALU clutter.
#define PF_DIST 256  // floats ahead on the features stream

__global__ void k_gemm(const float* __restrict__ xn,
                       const float* __restrict__ feats,
                       float* __restrict__ logits) {
  int lane  = threadIdx.x & 31;
  int wave  = threadIdx.x >> 5;
  int ntile = blockIdx.x * 8 + wave;          // 512 blocks * 8 waves = 4096 tiles
  int n0 = ntile * 16;
  int r  = lane & 15;
  int kk = lane >> 4;

  const float* bp = feats + (size_t)(n0 + r) * DD + 2 * kk;  // B: features rows
  const float* ap = xn + (size_t)r * DD + 2 * kk;            // A: x rows (tile 0)

  v8f acc0 = {}, acc1 = {}, acc2 = {}, acc3 = {};

#pragma unroll 4
  for (int k = 0; k < DD - PF_DIST; k += 4) {
    v2f bf = *(const v2f*)(bp + k);
    v2f a0 = *(const v2f*)(ap + k);
    v2f a1 = *(const v2f*)(ap + 16 * DD + k);
    v2f a2 = *(const v2f*)(ap + 32 * DD + k);
    v2f a3 = *(const v2f*)(ap + 48 * DD + k);
    __builtin_prefetch(bp + k + PF_DIST, 0, 1);   // global_prefetch_b8, in-bounds
    acc0 = __builtin_amdgcn_wmma_f32_16x16x4_f32(false, a0, false, bf, (short)0, acc0, false, false);
    acc1 = __builtin_amdgcn_wmma_f32_16x16x4_f32(false, a1, false, bf, (short)0, acc1, false, false);
    acc2 = __builtin_amdgcn_wmma_f32_16x16x4_f32(false, a2, false, bf, (short)0, acc2, false, false);
    acc3 = __builtin_amdgcn_wmma_f32_16x16x4_f32(false, a3, false, bf, (short)0, acc3, false, false);
  }
#pragma unroll 4
  for (int k = DD - PF_DIST; k < DD; k += 4) {
    v2f bf = *(const v2f*)(bp + k);
    v2f a0 = *(const v2f*)(ap + k);
    v2f a1 = *(const v2f*)(ap + 16 * DD + k);
    v2f a2 = *(const v2f*)(ap + 32 * DD + k);
    v2f a3 = *(const v2f*)(ap + 48 * DD + k);
    acc0 = __builtin_amdgcn_wmma_f32_16x16x4_f32(false, a0, false, bf, (short)0, acc0, false, false);
    acc1 = __builtin_amdgcn_wmma_f32_16x16x4_f32(false, a1, false, bf, (short)0, acc1, false, false);
    acc2 = __builtin_amdgcn_wmma_f32_16x16x4_f32(false, a2, false, bf, (short)0, acc2, false, false);
    acc3 = __builtin_amdgcn_wmma_f32_16x16x4_f32(false, a3, false, bf, (short)0, acc3, false, false);
  }

  // C/D layout: VGPR v, lanes0-15 -> M=v, lanes16-31 -> M=v+8; N = n0 + (lane&15)
  int n = n0 + r;
#pragma unroll
  for (int v = 0; v < 8; ++v) {
    int m = v + 8 * kk;
    logits[(size_t)(m)      * NN + n] = acc0[v];
    logits[(size_t)(m + 16) * NN + n] = acc1[v];
    logits[(size_t)(m + 32) * NN + n] = acc2[v];
    logits[(size_t)(m + 48) * NN + n] = acc3[v];
  }
}

// ---------------- K2: per-row max, CE loss term, entropy score ---------------
__global__ void k_rowstats(const float* __restrict__ logits,
                           const int* __restrict__ targets,
                           float* __restrict__ score,
                           float* __restrict__ lossterm) {
  __shared__ float red[256];
  int b = blockIdx.x, t = threadIdx.x;
  const float* row = logits + (size_t)b * NN;

  float m = -INFINITY;
  for (int i = t; i < NN; i += 256) m = fmaxf(m, row[i]);
  float M = blk_max(m, red);

  float sT = 0.f, sU = 0.f, wU = 0.f;
  for (int i = t; i < NN; i += 256) {
    float d = row[i] - M;
    sT += expf(d * T_INV);
    float u = d * U_INV;
    float e = expf(u);
    sU += e;
    wU += e * u;
  }
  float ST = blk_sum(sT, red);
  float SU = blk_sum(sU, red);
  float WU = blk_sum(wU, red);

  if (t == 0) {
    score[b] = logf(SU) - WU / SU;                         // entropy of softmax(l/0.5)
    float lt = row[targets[b]];
    lossterm[b] = -((lt - M) * T_INV - logf(ST));          // -log_softmax(l/0.05)[tgt]
  }
}

// ---------------- K3: stable ascending rank-sort, keep top-16 in order -------
__global__ void k_select(const float* __restrict__ score, int* __restrict__ unc) {
  int t = threadIdx.x;  // 64 threads
  float s = score[t];
  int rank = 0;
  for (int j = 0; j < BB; ++j) {
    float sj = score[j];
    rank += (sj < s) || (sj == s && j < t);   // stable, matches jnp.argsort
  }
  if (rank >= BB - UU) unc[rank - (BB - UU)] = t;
}

// ---------------- K4: per-uncertain-row softmax stats (TEMP == SIM_TEMP) -----
// lam = 1/S, CE term at argmax label = log S, with S = sum exp((l - max)/0.05)
__global__ void k_uncertain(const float* __restrict__ logits,
                            const int* __restrict__ unc,
                            float* __restrict__ lam_out,   // d_out + 2
                            float* __restrict__ uterm) {
  __shared__ float red[256];
  int j = blockIdx.x, t = threadIdx.x;
  const float* row = logits + (size_t)unc[j] * NN;

  float m = -INFINITY;
  for (int i = t; i < NN; i += 256) m = fmaxf(m, row[i]);
  float M = blk_max(m, red);

  float s = 0.f;
  for (int i = t; i < NN; i += 256) s += expf((row[i] - M) * T_INV);
  float S = blk_sum(s, red);

  if (t == 0) {
    lam_out[j] = 1.0f / S;
    uterm[j]   = logf(S);
  }
}

// ---------------- K5: scalar means (serial => deterministic) -----------------
__global__ void k_finalize(const float* __restrict__ lossterm,
                           const float* __restrict__ uterm,
                           float* __restrict__ out01) {
  if (threadIdx.x == 0) {
    float s = 0.f;
    for (int i = 0; i < BB; ++i) s += lossterm[i];
    out01[0] = s / (float)BB;
    float u = 0.f;
    for (int i = 0; i < UU; ++i) u += uterm[i];
    out01[1] = u / (float)UU;
  }
}

// ---------------- K6: copy features -> d_out+18 (8B-aligned, float2) ---------
__global__ void k_copy(const float* __restrict__ src, float* __restrict__ dst, long n2) {
  size_t i = (size_t)blockIdx.x * blockDim.x + threadIdx.x;
  size_t stride = (size_t)gridDim.x * blockDim.x;
  const v2f* s2 = (const v2f*)src;
  v2f* d2 = (v2f*)dst;
  for (; i < (size_t)n2; i += stride) d2[i] = s2[i];
}

// ---------------- K7: sequential momentum scan (single block) ----------------
// Element t of every row is only ever read/written by thread t, so the serial
// chain across duplicate targets is per-thread program order; only the norm
// reduction needs barriers.
__global__ void k_scan(float* __restrict__ nf,           // d_out + 18
                       const float* __restrict__ xn,
                       const int* __restrict__ targets) {
  __shared__ float red[1024];
  int t = threadIdx.x;  // 1024 threads, one per element
  for (int i = 0; i < BB; ++i) {
    int y = targets[i];
    float v = 0.2f * nf[(size_t)y * DD + t] + 0.8f * xn[(size_t)i * DD + t];
    red[t] = v * v; __syncthreads();
    for (int o = 512; o > 0; o >>= 1) {
      if (t < o) red[t] += red[t + o];
      __syncthreads();
    }
    float inv = 1.0f / fmaxf(sqrtf(red[0]), 1e-12f);
    __syncthreads();  // red[] reused next iteration
    nf[(size_t)y * DD + t] = v * inv;
  }
}

// ---------------------------------------------------------------------------
extern "C" void kernel_launch(void* const* d_in, const int* in_sizes, int n_in,
                              void* d_out, int out_size, void* d_ws, size_t ws_size,
                              hipStream_t stream) {
  const float* inputs   = (const float*)d_in[0];
  const float* features = (const float*)d_in[1];
  const int*   targets  = (const int*)d_in[2];
  // d_in[3] = uncertain_num (fixed to 16 per reference shapes)

  float* out = (float*)d_out;   // [loss][uloss][lam x16][new_features 64M]
  float* wsf = (float*)d_ws;

  float* xn       = wsf;                       // 64*1024
  float* logits   = wsf + 65536;               // 64*65536
  float* score    = logits + (size_t)BB * NN;  // 64
  float* lossterm = score + BB;                // 64
  float* uterm    = lossterm + BB;             // 16
  int*   unc      = (int*)(uterm + UU);        // 16

  k_norm_x  <<<BB,   256, 0, stream>>>(inputs, xn);
  k_gemm    <<<512,  256, 0, stream>>>(xn, features, logits);
  k_rowstats<<<BB,   256, 0, stream>>>(logits, targets, score, lossterm);
  k_select  <<<1,     64, 0, stream>>>(score, unc);
  k_uncertain<<<UU,  256, 0, stream>>>(logits, unc, out + 2, uterm);
  k_finalize<<<1,     32, 0, stream>>>(lossterm, uterm, out);
  k_copy    <<<8192, 256, 0, stream>>>(features, out + 18, (long)((size_t)NN * DD / 2));
  k_scan    <<<1,   1024, 0, stream>>>(out + 18, xn, targets);
}